// Bridge_2d_52312701665835
// MI455X (gfx1250) — compile-verified
//
#include <hip/hip_runtime.h>

typedef __attribute__((ext_vector_type(2))) float v2f;
typedef __attribute__((ext_vector_type(4))) float v4f;
typedef __attribute__((ext_vector_type(8))) float v8f;

#define HDIM   128
#define BATCH  4096
#define NTS    8
#define NWAVES 2
#define TILES  (BATCH / 16)          // 256 row-tiles of 16
#define NBLK   (TILES / NWAVES)      // 128 workgroups

// per-wave scratch region (floats)
#define PW_BUF   0            // 6 channels x 16x128    (12288)
#define PW_TMP   12288        // 2 x 16x128 (p0,p1)     (4096)
#define PW_L3    16384        // 6 x 16 x 2             (192)
#define PW_RSX0  16576
#define PW_RSX1  16592
#define PW_RSS0  16608
#define PW_RSS1  16624
#define PW_Y     16640        // 4 comps x 16
#define PW_K     16704        // 4 comps x 16
#define PW_YN    16768        // 4 comps x 16
#define PW_LOG   16832        // 16
#define PW_PEN   16848        // 16
#define PERWAVE  16896

// CDNA5 has a native V_TANH_F32 trans op (co-executes with XDL WMMA).
// Use it if the toolchain exposes a builtin; else exp + single v_rcp_f32
// (avoid the ~8-op IEEE division expansion the compiler emits for '/').
__device__ __forceinline__ float fast_tanh(float x) {
#if __has_builtin(__builtin_amdgcn_tanhf)
  return __builtin_amdgcn_tanhf(x);
#elif __has_builtin(__builtin_amdgcn_tanh_f32)
  return __builtin_amdgcn_tanh_f32(x);
#else
  float ax = __builtin_fabsf(x);
  float e  = __expf(2.0f * ax);                       // v_exp_f32
  float r  = 1.0f - 2.0f * __builtin_amdgcn_rcpf(e + 1.0f);  // v_rcp_f32
  return (x < 0.0f) ? -r : r;
#endif
}

// C = A(16x128, row-major in LDS) * W(128x128, pair-swizzled in LDS), f32 WMMA.
// A fragment (16x4): lane l -> row M=l&15, K pair base 2*(l>>4): contiguous b64.
// B fragment (4x16): lane l -> col N=l&15, same K pair: contiguous b64 thanks to
// the (k,k+1)-pair swizzle W_swz[(k>>1)*256 + n*2 + (k&1)].
// All 8 B fragments are preloaded so the ds_loads issue back-to-back and the
// WMMA burst drains them with decrementing dscnt waits.
__device__ __forceinline__ void mm128(const float* A, const float* W,
                                      v8f acc[8], int lo, int hi) {
#pragma unroll 1
  for (int kb = 0; kb < 32; ++kb) {
    v2f a = *(const v2f*)(A + (lo << 7) + (kb << 2) + (hi << 1));
    const float* wp = W + (((kb << 1) + hi) << 8) + (lo << 1);
    v2f b[8];
#pragma unroll
    for (int ct = 0; ct < 8; ++ct) b[ct] = *(const v2f*)(wp + (ct << 5));
#pragma unroll
    for (int ct = 0; ct < 8; ++ct)
      acc[ct] = __builtin_amdgcn_wmma_f32_16x16x4_f32(
          false, a, false, b[ct], (short)0, acc[ct], false, false);
  }
}

// Hidden layer, forward (z channel only), in place on buf.
__device__ __forceinline__ void hidden_fwd(float* buf, const float* W,
                                           const float* bias, int lo, int hi) {
  v8f acc[8] = {};
  mm128(buf, W, acc, lo, hi);
#pragma unroll
  for (int ct = 0; ct < 8; ++ct) {
    int n = (ct << 4) + lo;
    float bb = bias[n];
#pragma unroll
    for (int r = 0; r < 8; ++r) {
      int m = r + (hi << 3);
      buf[m * HDIM + n] = fast_tanh(acc[ct][r] + bb);
    }
  }
}

// Hidden layer, backward: 6 channels {z,u0,u1,v00,v01,v11}, each in place.
// tanh mixing:  z=tanh(p); d=1-z^2; u_a=d*p_a; v_ab=d*p_ab - 2 z d p_a p_b.
__device__ __forceinline__ void hidden_bwd(float* buf, float* tmp, const float* W,
                                           const float* bias, int lo, int hi) {
  {  // channel 0: z
    v8f acc[8] = {};
    mm128(buf, W, acc, lo, hi);
#pragma unroll
    for (int ct = 0; ct < 8; ++ct) {
      int n = (ct << 4) + lo;
      float bb = bias[n];
#pragma unroll
      for (int r = 0; r < 8; ++r) {
        int m = r + (hi << 3);
        buf[m * HDIM + n] = fast_tanh(acc[ct][r] + bb);
      }
    }
  }
#pragma unroll 1
  for (int a = 0; a < 2; ++a) {  // channels 1,2: first-order tangents
    v8f acc[8] = {};
    mm128(buf + (1 + a) * 2048, W, acc, lo, hi);
#pragma unroll
    for (int ct = 0; ct < 8; ++ct) {
      int n = (ct << 4) + lo;
#pragma unroll
      for (int r = 0; r < 8; ++r) {
        int m = r + (hi << 3), idx = m * HDIM + n;
        float z = buf[idx], dd = 1.0f - z * z;
        float p = acc[ct][r];
        tmp[a * 2048 + idx] = p;               // keep pre-act tangent for v mixing
        buf[(1 + a) * 2048 + idx] = dd * p;
      }
    }
  }
#pragma unroll 1
  for (int c = 0; c < 3; ++c) {  // channels 3,4,5: second order (00,01,11)
    int aa = (c == 2) ? 1 : 0;
    int bb2 = (c == 0) ? 0 : 1;
    v8f acc[8] = {};
    mm128(buf + (3 + c) * 2048, W, acc, lo, hi);
#pragma unroll
    for (int ct = 0; ct < 8; ++ct) {
      int n = (ct << 4) + lo;
#pragma unroll
      for (int r = 0; r < 8; ++r) {
        int m = r + (hi << 3), idx = m * HDIM + n;
        float z = buf[idx], dd = 1.0f - z * z;
        float pa = tmp[aa * 2048 + idx], pb = tmp[bb2 * 2048 + idx];
        buf[(3 + c) * 2048 + idx] = dd * acc[ct][r] - 2.0f * z * dd * pa * pb;
      }
    }
  }
}

// Input layer (K=3, VALU). Channels seeded analytically:
// u_a = d*W0[a], v_ab = -2 z d W0[a] W0[b].
__device__ __forceinline__ void input_layer(float* buf, const float* w0s,
                                            const float* b0s, const float* rx0,
                                            const float* rx1, float t,
                                            int lo, int hi, bool bwd) {
  float xs0[8], xs1[8];
#pragma unroll
  for (int r = 0; r < 8; ++r) {
    int m = r + (hi << 3);
    xs0[r] = rx0[m];
    xs1[r] = rx1[m];
  }
#pragma unroll 1
  for (int ct = 0; ct < 8; ++ct) {
    int n = (ct << 4) + lo;
    float wa = w0s[n], wb = w0s[HDIM + n], wt = w0s[2 * HDIM + n], bb = b0s[n];
#pragma unroll
    for (int r = 0; r < 8; ++r) {
      int m = r + (hi << 3), idx = m * HDIM + n;
      float p = fmaf(xs0[r], wa, fmaf(xs1[r], wb, fmaf(t, wt, bb)));
      float z = fast_tanh(p);
      buf[idx] = z;
      if (bwd) {
        float dd = 1.0f - z * z;
        buf[2048 + idx]     = dd * wa;
        buf[2 * 2048 + idx] = dd * wb;
        float c2 = -2.0f * z * dd;
        buf[3 * 2048 + idx] = c2 * wa * wa;
        buf[4 * 2048 + idx] = c2 * wa * wb;
        buf[5 * 2048 + idx] = c2 * wb * wb;
      }
    }
  }
}

// Output layer (128 -> 2). 32 lanes = 16 rows x 2 outputs; float4 LDS reads.
__device__ __forceinline__ void out_layer(const float* buf, const float* w3t,
                                          const float* b3s, float* l3,
                                          int lane, int nch) {
  int m = lane & 15, o = lane >> 4;
#pragma unroll 1
  for (int c = 0; c < nch; ++c) {
    const float* zr = buf + c * 2048 + m * HDIM;
    const float* wc = w3t + o * HDIM;
    float a = (c == 0) ? b3s[o] : 0.0f;
#pragma unroll 1
    for (int q = 0; q < 32; ++q) {
      v4f zz = *(const v4f*)(zr + (q << 2));
      v4f ww = *(const v4f*)(wc + (q << 2));
      a = fmaf(zz.x, ww.x, fmaf(zz.y, ww.y, fmaf(zz.z, ww.z, fmaf(zz.w, ww.w, a))));
    }
    l3[c * 32 + m * 2 + o] = a;
  }
}

__global__ void __launch_bounds__(32 * NWAVES)
cnf_kernel(const float* __restrict__ xg,
           const float* __restrict__ w0g, const float* __restrict__ b0g,
           const float* __restrict__ w1g, const float* __restrict__ b1g,
           const float* __restrict__ w2g, const float* __restrict__ b2g,
           const float* __restrict__ w3g, const float* __restrict__ b3g,
           float* __restrict__ out, float* __restrict__ wspen) {
  // Static LDS (~270 KB of the 320 KB WGP pool) so the compiler keeps every
  // access in addrspace(3): ds_load/ds_store with immediate offsets, no
  // generic-pointer null checks, no 64-bit address ALU.
  __shared__ __align__(16) float sw1[HDIM * HDIM];   // pair-swizzled
  __shared__ __align__(16) float sw2[HDIM * HDIM];   // pair-swizzled
  __shared__ __align__(16) float sw0[3 * HDIM];
  __shared__ __align__(16) float sb0[HDIM];
  __shared__ __align__(16) float sb1[HDIM];
  __shared__ __align__(16) float sb2[HDIM];
  __shared__ __align__(16) float w3t[2 * HDIM];
  __shared__ __align__(16) float sb3[8];
  __shared__ __align__(16) float swave[NWAVES][PERWAVE];

  int tid = threadIdx.x;
  __builtin_prefetch(w1g + tid * 64, 0, 0);   // global_prefetch_b8: warm L2 path
  __builtin_prefetch(w2g + tid * 64, 0, 0);

  // Stage weights once per workgroup. w1/w2 pair-swizzled so a B fragment
  // {W[k][n], W[k+1][n]} is one contiguous ds_load_b64.
  for (int i = tid; i < HDIM * HDIM; i += 32 * NWAVES) {
    int k = i >> 7, n = i & 127;
    int d = ((k >> 1) << 8) + (n << 1) + (k & 1);
    sw1[d] = w1g[i];
    sw2[d] = w2g[i];
  }
  for (int i = tid; i < 3 * HDIM; i += 32 * NWAVES) sw0[i] = w0g[i];
  for (int i = tid; i < HDIM; i += 32 * NWAVES) {
    sb0[i] = b0g[i]; sb1[i] = b1g[i]; sb2[i] = b2g[i];
  }
  for (int i = tid; i < 2 * HDIM; i += 32 * NWAVES)
    w3t[(i & 1) * HDIM + (i >> 1)] = w3g[i];   // w3t[o][k] = w3[k][o]
  if (tid < 2) sb3[tid] = b3g[tid];
  __syncthreads();   // only cross-wave sync needed; everything below is wave-private

  int wave = tid >> 5, lane = tid & 31, lo = lane & 15, hi = lane >> 4;
  float* wb   = &swave[wave][0];
  float* buf  = wb + PW_BUF;
  float* tmp  = wb + PW_TMP;
  float* l3   = wb + PW_L3;
  float* rx0  = wb + PW_RSX0;
  float* rx1  = wb + PW_RSX1;
  float* rs0  = wb + PW_RSS0;
  float* rs1  = wb + PW_RSS1;
  float* y    = wb + PW_Y;
  float* kpr  = wb + PW_K;
  float* yn   = wb + PW_YN;
  float* lacc = wb + PW_LOG;
  float* pacc = wb + PW_PEN;

  int tile  = blockIdx.x * NWAVES + wave;
  int grow0 = tile * 16;
  const float hstep = 1.0f / (float)NTS;

  if (lane < 16) {
    int m = lane;
    y[m]      = xg[(grow0 + m) * 2 + 0];
    y[16 + m] = xg[(grow0 + m) * 2 + 1];
  }

  // ---------------- forward RK4: dz/dt = f(z,t) ----------------
  for (int step = 0; step < NTS; ++step) {
    float t0 = hstep * (float)step, hh = hstep;
    if (lane < 16) { int m = lane; yn[m] = y[m]; yn[16 + m] = y[16 + m]; }
    for (int st = 0; st < 4; ++st) {
      float cs  = (st == 0) ? 0.0f : ((st == 3) ? 1.0f : 0.5f);
      float wgt = (st == 0 || st == 3) ? (1.0f / 6.0f) : (1.0f / 3.0f);
      if (lane < 16) {
        int m = lane;
        rx0[m] = y[m]      + ((st == 0) ? 0.0f : cs * kpr[m]);
        rx1[m] = y[16 + m] + ((st == 0) ? 0.0f : cs * kpr[16 + m]);
      }
      float ts = t0 + cs * hh;
      input_layer(buf, sw0, sb0, rx0, rx1, ts, lo, hi, false);
      hidden_fwd(buf, sw1, sb1, lo, hi);
      hidden_fwd(buf, sw2, sb2, lo, hi);
      out_layer(buf, w3t, sb3, l3, lane, 1);
      if (lane < 16) {
        int m = lane;
        float k0 = hh * l3[m * 2 + 0], k1 = hh * l3[m * 2 + 1];
        kpr[m] = k0; kpr[16 + m] = k1;
        yn[m] += wgt * k0; yn[16 + m] += wgt * k1;
      }
    }
    if (lane < 16) { int m = lane; y[m] = yn[m]; y[16 + m] = yn[16 + m]; }
  }

  // ---------------- terminal conditions ----------------
  if (lane < 16) {
    int m = lane;
    float z0 = y[m], z1 = y[16 + m];
    y[32 + m] = -z0;            // score
    y[48 + m] = -z1;
    lacc[m] = -1.8378770664093453f - 0.5f * (z0 * z0 + z1 * z1);  // -log(2pi) - |z|^2/2
    pacc[m] = 0.0f;
  }

  // ---------------- backward RK4 on (x, score); log/pen accumulate ----------------
  for (int step = 0; step < NTS; ++step) {
    float t0 = 1.0f - hstep * (float)step, hh = -hstep;
    if (lane < 16) {
      int m = lane;
#pragma unroll
      for (int c = 0; c < 4; ++c) yn[c * 16 + m] = y[c * 16 + m];
    }
    for (int st = 0; st < 4; ++st) {
      float cs  = (st == 0) ? 0.0f : ((st == 3) ? 1.0f : 0.5f);
      float wgt = (st == 0 || st == 3) ? (1.0f / 6.0f) : (1.0f / 3.0f);
      if (lane < 16) {
        int m = lane;
        float a = (st == 0) ? 0.0f : cs;
        rx0[m] = y[m]      + a * kpr[m];
        rx1[m] = y[16 + m] + a * kpr[16 + m];
        rs0[m] = y[32 + m] + a * kpr[32 + m];
        rs1[m] = y[48 + m] + a * kpr[48 + m];
      }
      float ts = t0 + cs * hh;
      input_layer(buf, sw0, sb0, rx0, rx1, ts, lo, hi, true);
      hidden_bwd(buf, tmp, sw1, sb1, lo, hi);
      hidden_bwd(buf, tmp, sw2, sb2, lo, hi);
      out_layer(buf, w3t, sb3, l3, lane, 6);
      if (lane < 16) {
        int m = lane;
        float f0  = l3[m * 2 + 0],       f1  = l3[m * 2 + 1];
        float j00 = l3[32 + m * 2 + 0],  j10 = l3[32 + m * 2 + 1];  // d f / d x0
        float j01 = l3[64 + m * 2 + 0],  j11 = l3[64 + m * 2 + 1];  // d f / d x1
        float a00_0 = l3[96 + m * 2 + 0],  a00_1 = l3[96 + m * 2 + 1];
        float a01_0 = l3[128 + m * 2 + 0], a01_1 = l3[128 + m * 2 + 1];
        float a11_0 = l3[160 + m * 2 + 0], a11_1 = l3[160 + m * 2 + 1];
        float div = j00 + j11;
        float gd0 = a00_0 + a01_1;       // grad(div) via symmetric v_ab
        float gd1 = a01_0 + a11_1;
        float s0 = rs0[m], s1 = rs1[m];
        float Js0 = j00 * s0 + j01 * s1;
        float Js1 = j10 * s0 + j11 * s1;
        float k0 = hh * f0, k1 = hh * f1;
        float k2 = hh * (-gd0 - Js0), k3 = hh * (-gd1 - Js1);
        kpr[m] = k0; kpr[16 + m] = k1; kpr[32 + m] = k2; kpr[48 + m] = k3;
        yn[m] += wgt * k0; yn[16 + m] += wgt * k1;
        yn[32 + m] += wgt * k2; yn[48 + m] += wgt * k3;
        lacc[m] += wgt * hh * (-div);
        float e0 = f0 + s0, e1 = f1 + s1;
        pacc[m] += wgt * hh * (e0 * e0 + e1 * e1);
      }
    }
    if (lane < 16) {
      int m = lane;
#pragma unroll
      for (int c = 0; c < 4; ++c) y[c * 16 + m] = yn[c * 16 + m];
    }
  }

  // ---------------- outputs ----------------
  if (lane < 16) out[1 + grow0 + lane] = lacc[lane];
  if (lane == 0) {
    float s = 0.0f;
    for (int i = 0; i < 16; ++i) s += pacc[i];
    wspen[tile] = s;
  }
}

// Deterministic final reductions (fixed-order trees; no float atomics).
__global__ void cnf_finalize(float* __restrict__ out, const float* __restrict__ wspen) {
  __shared__ float sred[256];
  int tid = threadIdx.x;
  float s = 0.0f;
  for (int i = tid; i < BATCH; i += 256) s += out[1 + i];
  sred[tid] = s;
  __syncthreads();
  for (int off = 128; off > 0; off >>= 1) {
    if (tid < off) sred[tid] += sred[tid + off];
    __syncthreads();
  }
  if (tid == 0) out[0] = -10.0f * (sred[0] / (float)BATCH);
  __syncthreads();
  sred[tid] = wspen[tid];   // exactly TILES == 256 partials
  __syncthreads();
  for (int off = 128; off > 0; off >>= 1) {
    if (tid < off) sred[tid] += sred[tid + off];
    __syncthreads();
  }
  if (tid == 0) out[1 + BATCH] = -(sred[0] / (float)BATCH);
}

extern "C" void kernel_launch(void* const* d_in, const int* in_sizes, int n_in,
                              void* d_out, int out_size, void* d_ws, size_t ws_size,
                              hipStream_t stream) {
  const float* xg  = (const float*)d_in[0];
  const float* w0g = (const float*)d_in[1];
  const float* b0g = (const float*)d_in[2];
  const float* w1g = (const float*)d_in[3];
  const float* b1g = (const float*)d_in[4];
  const float* w2g = (const float*)d_in[5];
  const float* b2g = (const float*)d_in[6];
  const float* w3g = (const float*)d_in[7];
  const float* b3g = (const float*)d_in[8];
  float* out   = (float*)d_out;
  float* wspen = (float*)d_ws;

  cnf_kernel<<<NBLK, 32 * NWAVES, 0, stream>>>(xg, w0g, b0g, w1g, b1g,
                                               w2g, b2g, w3g, b3g, out, wspen);
  cnf_finalize<<<1, 256, 0, stream>>>(out, wspen);
}